// GNN_76338748719623
// MI455X (gfx1250) — compile-verified
//
#include <hip/hip_runtime.h>

typedef float v2f __attribute__((ext_vector_type(2)));
typedef float v8f __attribute__((ext_vector_type(8)));

#define H_FEAT 64

__global__ void fill_f32(float* __restrict__ p, float v, int n) {
    int i = blockIdx.x * blockDim.x + threadIdx.x;
    if (i < n) p[i] = v;
}

__global__ void deg_count(const int* __restrict__ dst, float* __restrict__ deg, int E) {
    int e = blockIdx.x * blockDim.x + threadIdx.x;
    if (e < E) atomicAdd(&deg[dst[e]], 1.0f);
}

__global__ void make_dinv(const float* __restrict__ deg, float* __restrict__ dinv, int n) {
    int i = blockIdx.x * blockDim.x + threadIdx.x;
    if (i < n) dinv[i] = rsqrtf(deg[i]);   // deg >= 1 always (self-loops)
}

// h[M x 64] = X[M x K] @ W[K x 64]; agg = h * dinv^2 (self-loop init).
// One block = 128 threads = 4 waves; block owns 16 rows, wave w owns cols [16w,16w+16).
// Exact tiling: M = 100000 = 6250*16, K in {128,64}. No divergence -> EXEC all ones.
__global__ void __launch_bounds__(128)
gemm16_wmma_f32(const float* __restrict__ X, const float* __restrict__ W,
                const float* __restrict__ dinv,
                float* __restrict__ h, float* __restrict__ agg, int K) {
    const int lane  = threadIdx.x & 31;
    const int wave  = threadIdx.x >> 5;
    const int baseM = blockIdx.x * 16;
    const int baseN = wave * 16;
    const int half  = lane >> 4;          // 0: K pair (k,k+1), 1: (k+2,k+3)
    const int l15   = lane & 15;
    const int row   = baseM + l15;        // A row held by this lane
    const int col   = baseN + l15;        // B/D column held by this lane

    v8f acc = {};
    for (int k = 0; k < K; k += 4) {
        const int kb = k + 2 * half;
        v2f a, b;
        a.x = X[(size_t)row * K + kb];
        a.y = X[(size_t)row * K + kb + 1];
        b.x = W[kb * H_FEAT + col];
        b.y = W[(kb + 1) * H_FEAT + col];
        // D = A(16x4,f32) * B(4x16,f32) + C(16x16,f32)
        acc = __builtin_amdgcn_wmma_f32_16x16x4_f32(false, a, false, b,
                                                    (short)0, acc, false, false);
    }
#pragma unroll
    for (int r = 0; r < 8; ++r) {
        const int orow = baseM + half * 8 + r;   // D: vgpr r -> M=r (lanes<16) / r+8 (lanes>=16)
        const float di = dinv[orow];
        const float v  = acc[r];
        h[(size_t)orow * H_FEAT + col]   = v;
        agg[(size_t)orow * H_FEAT + col] = v * di * di;
    }
}

// 64 threads per edge: lane f handles feature f. Gather from L2-resident h,
// atomic f32 scatter-add into agg.
__global__ void edge_scatter(const int* __restrict__ src, const int* __restrict__ dst,
                             const float* __restrict__ dinv,
                             const float* __restrict__ h,
                             float* __restrict__ agg, int E) {
    int t = blockIdx.x * blockDim.x + threadIdx.x;
    int e = t >> 6;
    int f = t & 63;
    if (e < E) {
        int s = src[e], d = dst[e];
        float nrm = dinv[s] * dinv[d];
        atomicAdd(&agg[(size_t)d * H_FEAT + f], h[(size_t)s * H_FEAT + f] * nrm);
    }
}

__global__ void bias_relu(const float* __restrict__ agg, const float* __restrict__ b,
                          float* __restrict__ act, int NH) {
    int i = blockIdx.x * blockDim.x + threadIdx.x;
    if (i < NH) act[i] = fmaxf(agg[i] + b[i & 63], 0.0f);
}

__global__ void pool_accum(const float* __restrict__ act, const int* __restrict__ batch,
                           float* __restrict__ pooled, float* __restrict__ cnt, int N) {
    int t = blockIdx.x * blockDim.x + threadIdx.x;
    int i = t >> 6, f = t & 63;
    if (i < N) {
        int g = batch[i];
        atomicAdd(&pooled[g * H_FEAT + f], act[(size_t)i * H_FEAT + f]);
        if (f == 0) atomicAdd(&cnt[g], 1.0f);
    }
}

__global__ void final_linear(const float* __restrict__ pooled, const float* __restrict__ cnt,
                             const float* __restrict__ Wl, const float* __restrict__ bl,
                             float* __restrict__ out, int G) {
    int t = blockIdx.x * blockDim.x + threadIdx.x;
    if (t < G * 2) {
        int g = t >> 1, c = t & 1;
        float inv = 1.0f / fmaxf(cnt[g], 1.0f);
        float s = 0.0f;
        for (int f = 0; f < H_FEAT; ++f)
            s += pooled[g * H_FEAT + f] * inv * Wl[f * 2 + c];
        out[t] = s + bl[c];
    }
}

extern "C" void kernel_launch(void* const* d_in, const int* in_sizes, int n_in,
                              void* d_out, int out_size, void* d_ws, size_t ws_size,
                              hipStream_t stream) {
    const float* x   = (const float*)d_in[0];
    const int*   ei  = (const int*)d_in[1];
    const int*   bat = (const int*)d_in[2];
    const float* W1  = (const float*)d_in[3];
    const float* b1  = (const float*)d_in[4];
    const float* W2  = (const float*)d_in[5];
    const float* b2  = (const float*)d_in[6];
    const float* W3  = (const float*)d_in[7];
    const float* b3  = (const float*)d_in[8];
    const float* Wl  = (const float*)d_in[9];
    const float* bl  = (const float*)d_in[10];
    float* out = (float*)d_out;

    const int N   = in_sizes[2];          // 100000
    const int Fin = in_sizes[0] / N;      // 128
    const int E   = in_sizes[1] / 2;      // 3.2M
    const int G   = 256;
    const int* src = ei;
    const int* dst = ei + E;

    char* ws = (char*)d_ws;
    size_t o = 0;
    float* act    = (float*)(ws + o); o += (size_t)N * H_FEAT * 4;
    float* hbuf   = (float*)(ws + o); o += (size_t)N * H_FEAT * 4;
    float* agg    = (float*)(ws + o); o += (size_t)N * H_FEAT * 4;
    float* deg    = (float*)(ws + o); o += (size_t)N * 4;
    float* dinv   = (float*)(ws + o); o += (size_t)N * 4;
    float* pooled = (float*)(ws + o); o += (size_t)G * H_FEAT * 4;
    float* cnt    = (float*)(ws + o); o += (size_t)G * 4;
    (void)ws_size; (void)n_in; (void)out_size;

    const int NH      = N * H_FEAT;
    const int mBlocks = N / 16;                       // exact: 6250
    const unsigned eBlocks  = (unsigned)(((long long)E * 64 + 255) / 256);
    const unsigned nhBlocks = (unsigned)((NH + 255) / 256);

    // Degrees + symmetric normalization (done once, reused by all layers).
    fill_f32<<<(N + 255) / 256, 256, 0, stream>>>(deg, 1.0f, N);
    deg_count<<<(E + 255) / 256, 256, 0, stream>>>(dst, deg, E);
    make_dinv<<<(N + 255) / 256, 256, 0, stream>>>(deg, dinv, N);

    // Layer 1
    gemm16_wmma_f32<<<mBlocks, 128, 0, stream>>>(x, W1, dinv, hbuf, agg, Fin);
    edge_scatter<<<eBlocks, 256, 0, stream>>>(src, dst, dinv, hbuf, agg, E);
    bias_relu<<<nhBlocks, 256, 0, stream>>>(agg, b1, act, NH);
    // Layer 2
    gemm16_wmma_f32<<<mBlocks, 128, 0, stream>>>(act, W2, dinv, hbuf, agg, H_FEAT);
    edge_scatter<<<eBlocks, 256, 0, stream>>>(src, dst, dinv, hbuf, agg, E);
    bias_relu<<<nhBlocks, 256, 0, stream>>>(agg, b2, act, NH);
    // Layer 3
    gemm16_wmma_f32<<<mBlocks, 128, 0, stream>>>(act, W3, dinv, hbuf, agg, H_FEAT);
    edge_scatter<<<eBlocks, 256, 0, stream>>>(src, dst, dinv, hbuf, agg, E);
    bias_relu<<<nhBlocks, 256, 0, stream>>>(agg, b3, act, NH);

    // Global mean pool + final linear
    fill_f32<<<(G * H_FEAT + 255) / 256, 256, 0, stream>>>(pooled, 0.0f, G * H_FEAT);
    fill_f32<<<1, 256, 0, stream>>>(cnt, 0.0f, G);
    pool_accum<<<nhBlocks, 256, 0, stream>>>(act, bat, pooled, cnt, N);
    final_linear<<<2, 256, 0, stream>>>(pooled, cnt, Wl, bl, out, G);
}